// Answerer_65592740544757
// MI455X (gfx1250) — compile-verified
//
#include <hip/hip_runtime.h>
#include <hip/hip_bf16.h>

typedef float v2f __attribute__((ext_vector_type(2)));
typedef float v4f __attribute__((ext_vector_type(4)));
typedef float v8f __attribute__((ext_vector_type(8)));

#define NEG_INF (-1e30f)
#define HDIM    1024                 // H fixed by the reference
#define NITER   (HDIM / 8)           // 128 K-blocks of 8
#define NLANES  32

// ---------------------------------------------------------------------------
// Kernel 1: QA head GEMV via V_WMMA_F32_16X16X4_F32.
//
// Each wave owns a 16-row tile of (B*S, H). K-order permuted so each lane's
// A fragment is one contiguous float4 (virtual K {0,1,2,3} = real
// {k0,k0+1,k0+4,k0+5} for WMMA#1, {+2,+3,+6,+7} for WMMA#2). The matching B
// operand (W columns in permuted order, zero for N>=2) is wave-invariant, so
// it is materialized ONCE per block into a 64 KB LDS table; the hot loop is
// then 1 global_load_b128 (A stream) + 1 ds_load_b128 (B) + 2 WMMAs.
// ---------------------------------------------------------------------------
__global__ __launch_bounds__(256) void qa_head_wmma(
    const float* __restrict__ X,     // (B*S, H) row-major
    const float* __restrict__ mask,  // (B*S)
    const float* __restrict__ W,     // (H, 2) row-major
    const float* __restrict__ bias,  // (2)
    float* __restrict__ start_out,   // (B*S)
    float* __restrict__ end_out)     // (B*S)
{
    __shared__ v4f Btab[NITER * NLANES];   // 128 iters * 32 lanes * 16B = 64 KB

    // ---- one-time cooperative fill of the pre-selected B operand table ----
    for (int e = threadIdx.x; e < NITER * NLANES; e += 256) {
        const int t  = e >> 5;             // iteration index
        const int l  = e & 31;             // lane
        const int nn = l & 15;             // N column
        const int ss = l >> 4;             // half select
        const int c0 = 8 * t + 4 * ss;     // this lane's real W row base
        v4f v = {0.f, 0.f, 0.f, 0.f};
        if (nn < 2) {
            v.x = W[2 * (c0 + 0) + nn];    // b1.x
            v.y = W[2 * (c0 + 1) + nn];    // b1.y
            v.z = W[2 * (c0 + 2) + nn];    // b2.x
            v.w = W[2 * (c0 + 3) + nn];    // b2.y
        }
        Btab[e] = v;
    }
    __syncthreads();

    const int lane = threadIdx.x & 31;
    const int wave = threadIdx.x >> 5;
    const int tile = blockIdx.x * 8 + wave;      // 16-row tile index
    const int n    = lane & 15;
    const int sel  = lane >> 4;
    const long row0 = (long)tile * 16;
    const float* xrow = X + (row0 + n) * (long)HDIM + 4 * sel;

    v8f acc = {0.f, 0.f, 0.f, 0.f, 0.f, 0.f, 0.f, 0.f};

    const v4f* bp = Btab + lane;
    for (int t = 0; t < NITER; ++t) {
        v4f f  = *(const v4f*)(xrow + 8 * t);    // global_load_b128, A stream
        v4f bb = bp[t * NLANES];                 // ds_load_b128, pre-selected B
        v2f a1 = { f.x, f.y };
        v2f b1 = { bb.x, bb.y };
        v2f a2 = { f.z, f.w };
        v2f b2 = { bb.z, bb.w };
        acc = __builtin_amdgcn_wmma_f32_16x16x4_f32(false, a1, false, b1,
                                                    (short)0, acc, false, false);
        acc = __builtin_amdgcn_wmma_f32_16x16x4_f32(false, a2, false, b2,
                                                    (short)0, acc, false, false);
    }

    // D layout: VGPR r -> row (r + 8*sel); N = n. Apply bias + ans_mask, scatter.
    if (n < 2) {
        const float bv = bias[n];
        #pragma unroll
        for (int r = 0; r < 8; ++r) {
            const long g = row0 + r + 8 * sel;   // flat (b*S + s)
            const float m = mask[g];
            const float v = acc[r] + bv;
            const float o = v * m + (1.f - m) * NEG_INF;
            if (n == 0) start_out[g] = o;
            else        end_out[g]   = o;
        }
    }
}

// ---------------------------------------------------------------------------
// Kernel 2: banded span top-5 per batch.
// Valid spans: {(i,j): i>=4, i<=j<=i+29} plus (1,1),(2,2),(3,3).
// Thread-local sorted top-5 (ties -> lower flat index, matching lax.top_k),
// then LDS tree merge of sorted 5-lists.
// ---------------------------------------------------------------------------
#define TOPK 5
#define TPB  256

__global__ __launch_bounds__(TPB) void topk_spans(
    const float* __restrict__ start, const float* __restrict__ end,
    float* __restrict__ outS, float* __restrict__ outE, int S)
{
    const int b   = blockIdx.x;
    const int tid = threadIdx.x;
    const float* sl = start + (long)b * S;
    const float* el = end   + (long)b * S;

    float lv[TOPK];
    int   li[TOPK];
    #pragma unroll
    for (int q = 0; q < TOPK; ++q) { lv[q] = -3.0e38f; li[q] = 0x7fffffff; }

    auto ins = [&](float v, int idx) {
        if (v > lv[TOPK - 1] || (v == lv[TOPK - 1] && idx < li[TOPK - 1])) {
            lv[TOPK - 1] = v; li[TOPK - 1] = idx;
            #pragma unroll
            for (int k = TOPK - 1; k > 0; --k) {
                if (lv[k] > lv[k - 1] || (lv[k] == lv[k - 1] && li[k] < li[k - 1])) {
                    float tv = lv[k]; lv[k] = lv[k - 1]; lv[k - 1] = tv;
                    int   ti = li[k]; li[k] = li[k - 1]; li[k - 1] = ti;
                } else break;
            }
        }
    };

    for (int i = 4 + tid; i < S; i += TPB) {
        const float si = sl[i];
        int jmax = i + 29; if (jmax > S - 1) jmax = S - 1;
        for (int j = i; j <= jmax; ++j) ins(si + el[j], i * S + j);
    }
    if (tid == 0) {
        for (int p = 1; p <= 3; ++p) ins(sl[p] + el[p], p * S + p);
    }

    __shared__ float sv[TPB * TOPK];
    __shared__ int   sx[TPB * TOPK];
    #pragma unroll
    for (int q = 0; q < TOPK; ++q) { sv[tid * TOPK + q] = lv[q]; sx[tid * TOPK + q] = li[q]; }
    __syncthreads();

    for (int s = TPB / 2; s >= 1; s >>= 1) {
        if (tid < s) {
            float av[TOPK], bv[TOPK], mv[TOPK];
            int   ai[TOPK], bi[TOPK], mi[TOPK];
            #pragma unroll
            for (int q = 0; q < TOPK; ++q) {
                av[q] = sv[tid * TOPK + q];       ai[q] = sx[tid * TOPK + q];
                bv[q] = sv[(tid + s) * TOPK + q]; bi[q] = sx[(tid + s) * TOPK + q];
            }
            int pa = 0, pb = 0;
            #pragma unroll
            for (int q = 0; q < TOPK; ++q) {
                bool takeA;
                if      (pb >= TOPK) takeA = true;
                else if (pa >= TOPK) takeA = false;
                else takeA = (av[pa] > bv[pb]) ||
                             (av[pa] == bv[pb] && ai[pa] <= bi[pb]);
                if (takeA) { mv[q] = av[pa]; mi[q] = ai[pa]; ++pa; }
                else       { mv[q] = bv[pb]; mi[q] = bi[pb]; ++pb; }
            }
            #pragma unroll
            for (int q = 0; q < TOPK; ++q) { sv[tid * TOPK + q] = mv[q]; sx[tid * TOPK + q] = mi[q]; }
        }
        __syncthreads();
    }

    if (tid == 0) {
        #pragma unroll
        for (int q = 0; q < TOPK; ++q) {
            const int idx = sx[q];
            outS[b * TOPK + q] = (float)(idx / S);
            outE[b * TOPK + q] = (float)(idx % S);
        }
    }
}

// ---------------------------------------------------------------------------
extern "C" void kernel_launch(void* const* d_in, const int* in_sizes, int n_in,
                              void* d_out, int out_size, void* d_ws, size_t ws_size,
                              hipStream_t stream) {
    const float* X    = (const float*)d_in[0];  // seq_hiddens (B,S,H)
    const float* mask = (const float*)d_in[1];  // ans_mask    (B,S)
    const float* W    = (const float*)d_in[2];  // W_qa        (H,2)
    const float* bias = (const float*)d_in[3];  // b_qa        (2)

    const int BS = in_sizes[1];       // B*S
    const int S  = 4096;
    const int B  = BS / S;

    float* start_out = (float*)d_out;           // (B,S)
    float* end_out   = start_out + BS;          // (B,S)
    float* topS      = end_out + BS;            // (B,5) as float
    float* topE      = topS + B * TOPK;         // (B,5) as float

    // 16 rows per wave, 8 waves per block -> 128 rows per block.
    qa_head_wmma<<<BS / 128, 256, 0, stream>>>(X, mask, W, bias,
                                               start_out, end_out);
    topk_spans<<<B, TPB, 0, stream>>>(start_out, end_out, topS, topE, S);
}